// GroupNorm_23665269801344
// MI455X (gfx1250) — compile-verified
//
#include <hip/hip_runtime.h>

// Problem constants (from reference)
#define CHN 256
#define GRP 32
#define CPG 8                     // channels per group
#define HWD 147456                // 384*384
#define CH_FLOATS (HWD * 2)       // floats per channel = 294912
#define CH_F4 (CH_FLOATS / 4)     // float4 per channel = 73728
#define BLOCKS_PER_CH 32
#define SLICE_F4 (CH_F4 / BLOCKS_PER_CH)   // 2304
#define TPB 256
#define ITERS (SLICE_F4 / TPB)    // 9
#define EPSV 1e-5f

// ws layout (floats):
//   [0 .. 511]   per-channel sums (Sx0, Sx1) x 256
//   [512 .. 607] per-group raw 2nd moments (M00, M01, M11) x 32
//   [608 .. 767] per-group params (w00, w01, w11, mu0, mu1) x 32
#define WS_CHSUM 0
#define WS_GM2   512
#define WS_GPRM  608

#if defined(__gfx1250__) && __has_builtin(__builtin_amdgcn_global_load_async_to_lds_b128)
#define USE_ASYNC 1
#else
#define USE_ASYNC 0
#endif

// native clang vector (required by __builtin_nontemporal_*)
typedef float vf4 __attribute__((ext_vector_type(4)));

__global__ void wh_zero(float* __restrict__ ws) {
    int i = blockIdx.x * blockDim.x + threadIdx.x;
    if (i < WS_GPRM) ws[i] = 0.0f;
}

__global__ __launch_bounds__(TPB) void wh_reduce(const float* __restrict__ x,
                                                 float* __restrict__ ws) {
    const int c  = blockIdx.x / BLOCKS_PER_CH;
    const int sl = blockIdx.x % BLOCKS_PER_CH;
    const int g  = c / CPG;
    const int tid = threadIdx.x;
    const float4* __restrict__ src =
        reinterpret_cast<const float4*>(x) + (size_t)c * CH_F4 + (size_t)sl * SLICE_F4;

    float s0 = 0.f, s1 = 0.f, m00 = 0.f, m01 = 0.f, m11 = 0.f;

#if USE_ASYNC
    // CDNA5 async-to-LDS pipeline: double-buffered GLOBAL_LOAD_ASYNC_TO_LDS_B128.
    // Each lane async-loads 16B into its own LDS slot, so only s_wait_asynccnt
    // (no barrier) is needed before the lane reads its slot back. Default (RT)
    // cache policy on purpose: this first pass primes L2 with x for the apply pass.
    __shared__ float4 buf[2][TPB];
    typedef int v4i __attribute__((vector_size(16)));
    typedef __attribute__((address_space(1))) v4i gv4i;
    typedef __attribute__((address_space(3))) v4i lv4i;

    __builtin_amdgcn_global_load_async_to_lds_b128((gv4i*)(src + tid),
                                                   (lv4i*)&buf[0][tid], 0, 0);
#pragma unroll
    for (int i = 0; i < ITERS; ++i) {
        const int cur = i & 1;
        if (i + 1 < ITERS) {
            __builtin_amdgcn_global_load_async_to_lds_b128(
                (gv4i*)(src + (size_t)(i + 1) * TPB + tid),
                (lv4i*)&buf[cur ^ 1][tid], 0, 0);
            asm volatile("s_wait_asynccnt 1" ::: "memory");   // current buffer ready
        } else {
            asm volatile("s_wait_asynccnt 0" ::: "memory");   // drain
        }
        float4 v = buf[cur][tid];
        s0 += v.x + v.z;
        s1 += v.y + v.w;
        m00 = fmaf(v.x, v.x, fmaf(v.z, v.z, m00));
        m01 = fmaf(v.x, v.y, fmaf(v.z, v.w, m01));
        m11 = fmaf(v.y, v.y, fmaf(v.w, v.w, m11));
    }
#else
#pragma unroll
    for (int i = 0; i < ITERS; ++i) {
        float4 v = src[(size_t)i * TPB + tid];
        s0 += v.x + v.z;
        s1 += v.y + v.w;
        m00 = fmaf(v.x, v.x, fmaf(v.z, v.z, m00));
        m01 = fmaf(v.x, v.y, fmaf(v.z, v.w, m01));
        m11 = fmaf(v.y, v.y, fmaf(v.w, v.w, m11));
    }
#endif

    // wave32 shuffle reduction
#pragma unroll
    for (int off = 16; off > 0; off >>= 1) {
        s0  += __shfl_down(s0, off);
        s1  += __shfl_down(s1, off);
        m00 += __shfl_down(m00, off);
        m01 += __shfl_down(m01, off);
        m11 += __shfl_down(m11, off);
    }

    __shared__ float part[TPB / 32][5];
    const int lane = tid & 31, wv = tid >> 5;
    if (lane == 0) {
        part[wv][0] = s0;  part[wv][1] = s1;
        part[wv][2] = m00; part[wv][3] = m01; part[wv][4] = m11;
    }
    __syncthreads();
    if (tid < 5) {
        float a = 0.f;
#pragma unroll
        for (int w = 0; w < TPB / 32; ++w) a += part[w][tid];
        if (tid < 2) atomicAdd(&ws[WS_CHSUM + c * 2 + tid], a);
        else         atomicAdd(&ws[WS_GM2 + g * 3 + (tid - 2)], a);
    }
}

// Closed-form (cov + eps*I)^(-1/2) for 2x2 SPD; equals V diag((l+eps)^-1/2) V^T.
__global__ void wh_finalize(float* __restrict__ ws) {
    const int g = threadIdx.x;
    if (g >= GRP) return;
    float S0 = 0.f, S1 = 0.f;
#pragma unroll
    for (int k = 0; k < CPG; ++k) {
        S0 += ws[WS_CHSUM + (g * CPG + k) * 2 + 0];
        S1 += ws[WS_CHSUM + (g * CPG + k) * 2 + 1];
    }
    const float invN = 1.0f / (float)(CPG * HWD);
    const float mu0 = S0 * invN, mu1 = S1 * invN;
    const float b00 = ws[WS_GM2 + g * 3 + 0] * invN - mu0 * mu0 + EPSV;
    const float b01 = ws[WS_GM2 + g * 3 + 1] * invN - mu0 * mu1;
    const float b11 = ws[WS_GM2 + g * 3 + 2] * invN - mu1 * mu1 + EPSV;
    const float t = b00 + b11;
    const float d = fmaxf(b00 * b11 - b01 * b01, 1e-30f);
    const float s = sqrtf(d);
    const float k2 = 1.0f / (s * sqrtf(t + 2.0f * s));
    float* p = ws + WS_GPRM + g * 5;
    p[0] = (b11 + s) * k2;   // w00
    p[1] = -b01 * k2;        // w01
    p[2] = (b00 + s) * k2;   // w11
    p[3] = mu0;
    p[4] = mu1;
}

__global__ __launch_bounds__(TPB) void wh_apply(const float* __restrict__ x,
                                                const float* __restrict__ scale,
                                                const float* __restrict__ bias,
                                                const float* __restrict__ ws,
                                                float* __restrict__ out) {
    // Traverse x in REVERSE block order: the reduce pass just streamed all 302MB
    // of x through the 192MB L2, so the tail of x is still resident. Consuming
    // it tail-first turns up to ~190MB of this second read into L2 hits.
    const int rb = (int)gridDim.x - 1 - (int)blockIdx.x;
    const int c  = rb / BLOCKS_PER_CH;
    const int sl = rb % BLOCKS_PER_CH;
    const int g  = c / CPG;
    const float* p = ws + WS_GPRM + g * 5;
    const float sc = scale[c], bi = bias[c];
    const float inv_hw = 1.0f / (float)HWD;
    const float mv0 = ws[WS_CHSUM + c * 2 + 0] * inv_hw;
    const float mv1 = ws[WS_CHSUM + c * 2 + 1] * inv_hw;
    const float a00 = p[0] * sc, a01 = p[1] * sc, a11 = p[2] * sc;
    const float mu0 = p[3], mu1 = p[4];
    // y = a * (x - mu) + bias*mean_vec  ->  y = a*x + off
    const float off0 = fmaf(bi, mv0, -(a00 * mu0 + a01 * mu1));
    const float off1 = fmaf(bi, mv1, -(a01 * mu0 + a11 * mu1));

    const size_t base = (size_t)c * CH_F4 + (size_t)sl * SLICE_F4;
    const vf4* __restrict__ src = reinterpret_cast<const vf4*>(x) + base;
    vf4* __restrict__ dst = reinterpret_cast<vf4*>(out) + base;

#pragma unroll
    for (int i = 0; i < ITERS; ++i) {
        const size_t idx = (size_t)i * TPB + threadIdx.x;
        // NT load: this line of x is dead after this read -- don't let a miss
        // evict the still-resident tail. NT store: the 302MB output stream must
        // not pollute L2 either.
        vf4 v = __builtin_nontemporal_load(src + idx);
        vf4 r;
        r.x = fmaf(a00, v.x, fmaf(a01, v.y, off0));
        r.y = fmaf(a01, v.x, fmaf(a11, v.y, off1));
        r.z = fmaf(a00, v.z, fmaf(a01, v.w, off0));
        r.w = fmaf(a01, v.z, fmaf(a11, v.w, off1));
        __builtin_nontemporal_store(r, dst + idx);
    }
}

extern "C" void kernel_launch(void* const* d_in, const int* in_sizes, int n_in,
                              void* d_out, int out_size, void* d_ws, size_t ws_size,
                              hipStream_t stream) {
    (void)in_sizes; (void)n_in; (void)out_size; (void)ws_size;
    const float* x     = (const float*)d_in[0];
    const float* scale = (const float*)d_in[1];
    const float* bias  = (const float*)d_in[2];
    float* out = (float*)d_out;
    float* ws  = (float*)d_ws;

    wh_zero<<<3, 256, 0, stream>>>(ws);                                  // re-zero stats each call
    wh_reduce<<<CHN * BLOCKS_PER_CH, TPB, 0, stream>>>(x, ws);           // one-pass moments (primes L2)
    wh_finalize<<<1, 32, 0, stream>>>(ws);                               // 2x2 inverse sqrt, closed form
    wh_apply<<<CHN * BLOCKS_PER_CH, TPB, 0, stream>>>(x, scale, bias, ws, out); // reverse-order, NT I/O
}